// CompositionalLoss_82033875353978
// MI455X (gfx1250) — compile-verified
//
#include <hip/hip_runtime.h>
#include <hip/hip_bf16.h>

// ---------------------------------------------------------------------------
// CompositionalLoss on MI455X (gfx1250, wave32).
// delta = [C | -T] (210x42, +/-1 entries) @ X (42 x B*3), out = sum|delta|/B.
// Memory-bound: 33MB read once (~1.4us @ 23.3TB/s). GEMM via
// V_WMMA_F32_16X16X4_F32 (exact f32 precision; A entries are +/-1).
// ---------------------------------------------------------------------------

typedef float v2f __attribute__((ext_vector_type(2)));
typedef float v8f __attribute__((ext_vector_type(8)));

#define NJ      21
#define KDIM    42            // [input joints | target joints]
#define K_PAD   44            // pad to multiple of 4
#define KSTEPS  11            // K_PAD / 4
#define NPAIRS  210
#define P_TILES 14            // ceil(210/16) -> 224 rows padded
#define P_PAD   224
#define FRAG_ELEMS (P_TILES * KSTEPS * 32 * 2)   // 9856 floats, 39424 B

__device__ __constant__ int c_par[NJ] = {
    3, 0, 12, 3, 11, 7, 4, 9, 1, 3, 5, 8, 1, 20, 16, 13, 18, 1, 3, 14, 17
};

// ---------------------------------------------------------------------------
// Init: build A = [C | -T] (exactly replicating the reference path rule) and
// store it pre-swizzled into WMMA 16x4 f32 A-fragment layout:
//   frag[((pt*11+ks)*32 + lane)*2 + vg] = A[pt*16 + lane%16][ks*4 + (lane<16?vg:vg+2)]
// ---------------------------------------------------------------------------
__global__ void cl_init_afrag(float* __restrict__ afrag) {
    __shared__ float Afull[P_PAD * K_PAD];
    const int tid = threadIdx.x;
    for (int i = tid; i < P_PAD * K_PAD; i += blockDim.x) Afull[i] = 0.0f;
    __syncthreads();

    if (tid < NPAIRS) {
        // pair index -> (u, v) in combinations order
        int u = 0, v = 1, acc = 0;
        for (int a = 0; a < NJ; a++) {
            int cnt = NJ - 1 - a;
            if (tid < acc + cnt) { u = a; v = a + 1 + (tid - acc); break; }
            acc += cnt;
        }
        // depths
        int du = 0, dv = 0, x;
        x = u; while (c_par[x] != x) { x = c_par[x]; du++; }
        x = v; while (c_par[x] != x) { x = c_par[x]; dv++; }
        // LCA
        int uu = u, vv = v, d1 = du, d2 = dv;
        while (d1 > d2) { uu = c_par[uu]; d1--; }
        while (d2 > d1) { vv = c_par[vv]; d2--; }
        while (uu != vv) { uu = c_par[uu]; vv = c_par[vv]; }
        const int lca = uu;
        // path u..lca..v
        int path[2 * NJ]; int plen = 0;
        x = u; while (x != lca) { path[plen++] = x; x = c_par[x]; }
        path[plen++] = lca;
        int tmp[NJ]; int tl = 0;
        x = v; while (x != lca) { tmp[tl++] = x; x = c_par[x]; }
        for (int i = tl - 1; i >= 0; i--) path[plen++] = tmp[i];
        // reference rule: C[pi, pa[m]] += (PAR[pa[m]] == pa[m+1]) ? +1 : -1
        float* row = &Afull[tid * K_PAD];
        for (int m = 0; m < plen - 1; m++) {
            if (c_par[path[m]] == path[m + 1]) row[path[m]] += 1.0f;
            else                               row[path[m]] -= 1.0f;
        }
        // A[:, 21+j] = -T[:, j];  T[pi,u]=+1, T[pi,v]=-1
        row[NJ + u] -= 1.0f;
        row[NJ + v] += 1.0f;
    }
    __syncthreads();

    for (int i = tid; i < FRAG_ELEMS; i += blockDim.x) {
        const int vg   = i & 1;
        const int lane = (i >> 1) & 31;
        const int pk   = i >> 6;          // pt*KSTEPS + ks
        const int ks   = pk % KSTEPS;
        const int pt   = pk / KSTEPS;
        const int row  = pt * 16 + (lane & 15);
        const int k    = ks * 4 + ((lane < 16) ? vg : (vg + 2));
        afrag[i] = Afull[row * K_PAD + k];
    }
}

// ---------------------------------------------------------------------------
// Main: each wave grid-strides over 16-column tiles (col = b*3 + c).
// B fragments gathered from global (X read exactly once -> HBM bound),
// A fragments streamed from LDS (ds_load_b64, conflict-free).
// ---------------------------------------------------------------------------
__global__ void __launch_bounds__(256)
cl_main(const float* __restrict__ input, const float* __restrict__ target,
        const float* __restrict__ afrag, float* __restrict__ partials,
        int ncols, int ntiles) {
    __shared__ float lds_a[FRAG_ELEMS];
    const int tid = threadIdx.x;
    for (int i = tid; i < FRAG_ELEMS; i += blockDim.x) lds_a[i] = afrag[i];
    __syncthreads();

    const int lane          = tid & 31;
    const int half          = lane >> 4;        // 0: K+{0,1}, 1: K+{2,3}
    const int nloc          = lane & 15;        // column within tile
    const int wavesPerBlock = blockDim.x >> 5;
    const int gwave         = blockIdx.x * wavesPerBlock + (tid >> 5);
    const int nwaves        = gridDim.x * wavesPerBlock;

    const v2f* lds_v2 = (const v2f*)lds_a;
    float lsum = 0.0f;

    for (int tile = gwave; tile < ntiles; tile += nwaves) {
        const int col   = tile * 16 + nloc;
        const bool okc  = (col < ncols);
        const int colc  = okc ? col : 0;
        const int b     = colc / 3;
        const int c     = colc - 3 * b;
        const float* ib = input  + b * (NJ * 3) + c;
        const float* tb = target + b * (NJ * 3) + c;

        // B fragments: 4x16 f32 tile per k-step; lane holds column nloc,
        // VGPR0 -> K = ks*4 + {0|2}, VGPR1 -> K = ks*4 + {1|3}.
        v2f bfrag[KSTEPS];
#pragma unroll
        for (int ks = 0; ks < KSTEPS; ks++) {
#pragma unroll
            for (int vg = 0; vg < 2; vg++) {
                const int k = ks * 4 + vg + half * 2;
                float val = 0.0f;
                if (okc) {
                    if (k < NJ)         val = ib[k * 3];
                    else if (k < KDIM)  val = tb[(k - NJ) * 3];
                }
                bfrag[ks][vg] = val;
            }
        }

#pragma unroll 2
        for (int pt = 0; pt < P_TILES; pt++) {
            v8f acc = {};
            const v2f* ap = lds_v2 + (pt * KSTEPS) * 32 + lane;
#pragma unroll
            for (int ks = 0; ks < KSTEPS; ks++) {
                const v2f a = ap[ks * 32];   // ds_load_b64, bank-conflict free
                acc = __builtin_amdgcn_wmma_f32_16x16x4_f32(
                    false, a, false, bfrag[ks], (short)0, acc, false, false);
            }
#pragma unroll
            for (int r = 0; r < 8; r++) lsum += __builtin_fabsf(acc[r]);
        }
    }

    // deterministic in-wave reduction (wave32)
#pragma unroll
    for (int off = 16; off > 0; off >>= 1) lsum += __shfl_xor(lsum, off, 32);
    if (lane == 0) partials[gwave] = lsum;
}

// ---------------------------------------------------------------------------
// Finalize: fixed-order reduction -> deterministic scalar.
// ---------------------------------------------------------------------------
__global__ void cl_finalize(const float* __restrict__ partials, int n,
                            float* __restrict__ out, float invB) {
    __shared__ float sh[256];
    float s = 0.0f;
    for (int i = threadIdx.x; i < n; i += 256) s += partials[i];
    sh[threadIdx.x] = s;
    __syncthreads();
    if (threadIdx.x == 0) {
        float t = 0.0f;
        for (int i = 0; i < 256; i++) t += sh[i];
        out[0] = t * invB;
    }
}

extern "C" void kernel_launch(void* const* d_in, const int* in_sizes, int n_in,
                              void* d_out, int out_size, void* d_ws, size_t ws_size,
                              hipStream_t stream) {
    const float* input  = (const float*)d_in[0];
    const float* target = (const float*)d_in[1];
    const int Bn     = in_sizes[0] / (NJ * 3);       // 65536
    const int ncols  = Bn * 3;                        // 196608
    const int ntiles = (ncols + 15) / 16;             // 12288

    float* afrag    = (float*)d_ws;
    float* partials = afrag + FRAG_ELEMS;

    const int threads = 256;                          // 8 waves/block
    const int blocks  = 256;
    const int nwaves  = blocks * (threads / 32);      // 2048 partials

    cl_init_afrag<<<1, 256, 0, stream>>>(afrag);
    cl_main<<<blocks, threads, 0, stream>>>(input, target, afrag, partials,
                                            ncols, ntiles);
    cl_finalize<<<1, 256, 0, stream>>>(partials, nwaves, (float*)d_out,
                                       1.0f / (float)Bn);
}